// DTN_47459388620856
// MI455X (gfx1250) — compile-verified
//
#include <hip/hip_runtime.h>
#include <math.h>

#define HEADS 4
#define RES   28
#define PS    14
#define PP    196        // PS*PS pooled patches
#define TTOK  784        // RES*RES tokens
#define CDIM  768
#define CH    192        // channels per head
#define NPAD  208        // PP padded to 13 tiles of 16 for WMMA N
#define EPSV  1e-5f

typedef __attribute__((ext_vector_type(2))) float v2f;
typedef __attribute__((ext_vector_type(8))) float v8f;

// ---------------------------------------------------------------------------
// K1: positional softmax.  pos[h][i][j] = softmax_i( rel(i,j) . pos_w[h] + b )
// Stored as B-matrix [h][K=196][N=208], padding columns zeroed.
// grid = HEADS blocks of 208 threads (one thread per output column j).
// ---------------------------------------------------------------------------
__global__ void dtn_pos_kernel(const float* __restrict__ pos_w,
                               const float* __restrict__ pos_b,
                               float* __restrict__ pos) {
    int h = blockIdx.x;
    int j = threadIdx.x;                       // 0..207
    float* base = pos + (size_t)h * PP * NPAD;
    if (j >= PP) {                             // zero the N padding
        for (int i = 0; i < PP; ++i) base[i * NPAD + j] = 0.0f;
        return;
    }
    float w0 = pos_w[h * 3 + 0], w1 = pos_w[h * 3 + 1], w2 = pos_w[h * 3 + 2];
    float pb = pos_b[h];
    int jx = j % PS, jy = j / PS;

    float m = -1e30f;
    for (int i = 0; i < PP; ++i) {
        float ix = (float)(jx - (i % PS));     // indx = (j%14) - (i%14)
        float iy = (float)(jy - (i / PS));     // indy = (j/14) - (i/14)
        float s  = w0 * ix + w1 * iy + w2 * (ix * ix + iy * iy) + pb;
        m = fmaxf(m, s);
    }
    float sum = 0.0f;
    for (int i = 0; i < PP; ++i) {
        float ix = (float)(jx - (i % PS));
        float iy = (float)(jy - (i / PS));
        float s  = w0 * ix + w1 * iy + w2 * (ix * ix + iy * iy) + pb;
        sum += expf(s - m);
    }
    float inv = 1.0f / sum;
    for (int i = 0; i < PP; ++i) {
        float ix = (float)(jx - (i % PS));
        float iy = (float)(jy - (i / PS));
        float s  = w0 * ix + w1 * iy + w2 * (ix * ix + iy * iy) + pb;
        base[i * NPAD + j] = expf(s - m) * inv;
    }
}

// ---------------------------------------------------------------------------
// K2: RMS-normalize, layer stats, and 2x2 pooling.
// One block per (b, pooled patch p): 512 threads = 16 waves,
// wave (dt,hh) = (token-in-pool, head).  Deterministic shfl_xor reductions.
// xn written into d_out; pooled xp laid out [b][p][c] (c fastest).
// ---------------------------------------------------------------------------
__global__ void dtn_norm_pool_kernel(const float* __restrict__ x,
                                     float* __restrict__ xn_out,
                                     float* __restrict__ xp,
                                     float* __restrict__ mean_ln,
                                     float* __restrict__ var_ln) {
    __shared__ float lds_xn[4][CDIM];
    __shared__ float ln1[4][4];
    __shared__ float ln2[4][4];

    int blk  = blockIdx.x;
    int b    = blk / PP;
    int p    = blk % PP;
    int py   = p / PS, px = p % PS;
    int tid  = threadIdx.x;
    int w    = tid >> 5;
    int lane = tid & 31;
    int dt   = w >> 2;                 // token within 2x2 pool
    int hh   = w & 3;                  // head
    int ty   = py * 2 + (dt >> 1);
    int tx   = px * 2 + (dt & 1);
    int t    = ty * RES + tx;
    size_t base = ((size_t)(b * TTOK + t)) * CDIM + (size_t)hh * CH;

    float v[6];
    float ss = 0.0f;
    #pragma unroll
    for (int k = 0; k < 6; ++k) {
        v[k] = x[base + lane + 32 * k];
        ss += v[k] * v[k];
    }
    #pragma unroll
    for (int off = 16; off; off >>= 1) ss += __shfl_xor(ss, off, 32);
    float rsq = rsqrtf(ss * (1.0f / CH) + EPSV);

    float s1 = 0.0f, s2 = 0.0f;
    #pragma unroll
    for (int k = 0; k < 6; ++k) {
        float xv = v[k] * rsq;
        xn_out[base + lane + 32 * k] = xv;
        lds_xn[dt][hh * CH + lane + 32 * k] = xv;
        s1 += xv;
        s2 += xv * xv;
    }
    #pragma unroll
    for (int off = 16; off; off >>= 1) {
        s1 += __shfl_xor(s1, off, 32);
        s2 += __shfl_xor(s2, off, 32);
    }
    if (lane == 0) { ln1[dt][hh] = s1; ln2[dt][hh] = s2; }
    __syncthreads();

    // 2x2 average pool -> xp[b][p][c]
    for (int c = tid; c < CDIM; c += 512) {
        float pv = 0.25f * (lds_xn[0][c] + lds_xn[1][c] + lds_xn[2][c] + lds_xn[3][c]);
        xp[((size_t)(b * PP + p)) * CDIM + c] = pv;
    }
    // per-token layer mean / unbiased variance
    if (tid < 4) {
        int dtt = tid;
        float a1 = ln1[dtt][0] + ln1[dtt][1] + ln1[dtt][2] + ln1[dtt][3];
        float a2 = ln2[dtt][0] + ln2[dtt][1] + ln2[dtt][2] + ln2[dtt][3];
        int tyy = py * 2 + (dtt >> 1);
        int txx = px * 2 + (dtt & 1);
        int tt  = tyy * RES + txx;
        mean_ln[b * TTOK + tt] = a1 * (1.0f / CDIM);
        var_ln[b * TTOK + tt]  = (a2 - a1 * a1 * (1.0f / CDIM)) * (1.0f / (CDIM - 1));
    }
}

// ---------------------------------------------------------------------------
// K3: WMMA f32 einsum.  D[c,j] = sum_i xp[b,h,c,i] * pos[h,i,j] (and xp^2).
// One wave per 16x16 tile: M tiles = 192/16 = 12, N tiles = 208/16 = 13,
// K = 196 in 49 steps of v_wmma_f32_16x16x4_f32.
// grid = 2496 blocks x 256 thr = 19968 waves = 32*4*12*13 exactly.
// ---------------------------------------------------------------------------
__global__ void dtn_einsum_wmma_kernel(const float* __restrict__ xp,
                                       const float* __restrict__ pos,
                                       float* __restrict__ meanp,
                                       float* __restrict__ mean2p) {
    int tid  = threadIdx.x;
    int lane = tid & 31;
    int wg   = blockIdx.x * 8 + (tid >> 5);
    int nt   = wg % 13;
    int mt   = (wg / 13) % 12;
    int h    = (wg / 156) % 4;
    int b    = wg / 624;

    int m    = lane & 15;                 // A: M = lane%16 ; B/D: N = lane%16
    int kb   = (lane >> 4) << 1;          // lanes 16-31 hold K+2,K+3
    int crow = mt * 16 + m;               // channel row within head
    int jcol = nt * 16 + m;               // output pooled-patch column

    const float* Abase = xp  + (size_t)b * PP * CDIM + (size_t)h * CH + crow;
    const float* Bbase = pos + (size_t)h * PP * NPAD + jcol;

    v8f acc1 = {0.f, 0.f, 0.f, 0.f, 0.f, 0.f, 0.f, 0.f};
    v8f acc2 = {0.f, 0.f, 0.f, 0.f, 0.f, 0.f, 0.f, 0.f};

    for (int k0 = 0; k0 < PP; k0 += 4) {
        int k = k0 + kb;
        v2f a, a2, bb;
        a.x  = Abase[(size_t)k * CDIM];
        a.y  = Abase[(size_t)(k + 1) * CDIM];
        bb.x = Bbase[(size_t)k * NPAD];
        bb.y = Bbase[(size_t)(k + 1) * NPAD];
        a2.x = a.x * a.x;
        a2.y = a.y * a.y;
        acc1 = __builtin_amdgcn_wmma_f32_16x16x4_f32(false, a,  false, bb,
                                                     (short)0, acc1, false, false);
        acc2 = __builtin_amdgcn_wmma_f32_16x16x4_f32(false, a2, false, bb,
                                                     (short)0, acc2, false, false);
    }

    int rowadd = (lane >> 4) << 3;        // lanes 16-31 hold rows M+8
    if (jcol < PP) {
        #pragma unroll
        for (int r = 0; r < 8; ++r) {
            int c = mt * 16 + r + rowadd;
            size_t o = ((size_t)(b * PP + jcol)) * CDIM + (size_t)h * CH + c;
            meanp[o]  = acc1[r];
            mean2p[o] = acc2[r];
        }
    }
}

// ---------------------------------------------------------------------------
// K4: unpool + sigmoid blend + final normalize + affine, in place on d_out.
// One block per (b,t), coalesced over c.
// ---------------------------------------------------------------------------
__global__ void dtn_final_kernel(const float* __restrict__ weight,
                                 const float* __restrict__ bias,
                                 const float* __restrict__ mnw,
                                 const float* __restrict__ vnw,
                                 const float* __restrict__ meanp,
                                 const float* __restrict__ mean2p,
                                 const float* __restrict__ mean_ln,
                                 const float* __restrict__ var_ln,
                                 float* __restrict__ out) {
    int blk = blockIdx.x;
    int b = blk / TTOK, t = blk % TTOK;
    int ty = t / RES, tx = t % RES;
    int p  = (ty >> 1) * PS + (tx >> 1);    // nearest unpool source
    float ml = mean_ln[b * TTOK + t];
    float vl = var_ln[b * TTOK + t];
    size_t obase = ((size_t)(b * TTOK + t)) * CDIM;
    size_t pbase = ((size_t)(b * PP + p)) * CDIM;

    for (int c = threadIdx.x; c < CDIM; c += 256) {
        int h = c / CH;
        float mw = 1.0f / (1.0f + expf(-mnw[h]));
        float vw = 1.0f / (1.0f + expf(-vnw[h]));
        float mp  = meanp[pbase + c];
        float m2p = mean2p[pbase + c];
        float vp  = m2p - mp * mp;
        float mr  = (1.0f - mw) * mp + mw * ml;
        float vr  = (1.0f - vw) * vp + vw * vl;
        float xv  = out[obase + c];
        out[obase + c] = (xv - mr) * rsqrtf(vr + EPSV) * weight[c] + bias[c];
    }
}

// ---------------------------------------------------------------------------
// launcher
// ---------------------------------------------------------------------------
extern "C" void kernel_launch(void* const* d_in, const int* in_sizes, int n_in,
                              void* d_out, int out_size, void* d_ws, size_t ws_size,
                              hipStream_t stream) {
    const float* x     = (const float*)d_in[0];
    const float* wgt   = (const float*)d_in[1];
    const float* bia   = (const float*)d_in[2];
    const float* mnw   = (const float*)d_in[3];
    const float* vnw   = (const float*)d_in[4];
    const float* pos_w = (const float*)d_in[5];
    const float* pos_b = (const float*)d_in[6];
    float* out = (float*)d_out;

    const int B = 32;
    // workspace layout (floats)
    const size_t XP_N   = (size_t)B * PP * CDIM;      // 4,816,896
    float* ws       = (float*)d_ws;
    float* xp       = ws;                              // [b][p][c]
    float* meanp    = xp + XP_N;                       // [b][p][c]
    float* mean2p   = meanp + XP_N;                    // [b][p][c]
    float* mean_ln  = mean2p + XP_N;                   // [b][t]
    float* var_ln   = mean_ln + (size_t)B * TTOK;
    float* pos      = var_ln + (size_t)B * TTOK;       // [h][196][208]

    (void)in_sizes; (void)n_in; (void)out_size; (void)ws_size;

    dtn_pos_kernel<<<HEADS, NPAD, 0, stream>>>(pos_w, pos_b, pos);
    dtn_norm_pool_kernel<<<B * PP, 512, 0, stream>>>(x, out, xp, mean_ln, var_ln);
    dtn_einsum_wmma_kernel<<<(B * HEADS * 12 * 13) / 8, 256, 0, stream>>>(xp, pos, meanp, mean2p);
    dtn_final_kernel<<<B * TTOK, 256, 0, stream>>>(wgt, bia, mnw, vnw,
                                                   meanp, mean2p, mean_ln, var_ln, out);
}